// PokemonGraphEncoder_7911329759934
// MI455X (gfx1250) — compile-verified
//
#include <hip/hip_runtime.h>
#include <hip/hip_bf16.h>

typedef __attribute__((ext_vector_type(2))) float v2f;
typedef __attribute__((ext_vector_type(8))) float v8f;

#define N_NODES 100000
#define N_EDGES 800000
#define IN_DIM  64
#define HID_DIM 64
#define OUT_DIM 128
#define N_GRAPHS 64
#define LN_EPS 1e-5f

// ---------------------------------------------------------------------------
// Degree / coefficient precompute
// ---------------------------------------------------------------------------
__global__ void init_deg_kernel(float* deg, int n) {
    int i = blockIdx.x * blockDim.x + threadIdx.x;
    if (i < n) deg[i] = 1.0f;  // self-loop counted at target
}

__global__ void deg_edges_kernel(const long long* __restrict__ dst, float* deg, int e) {
    int i = blockIdx.x * blockDim.x + threadIdx.x;
    if (i < e) atomicAdd(&deg[(int)dst[i]], 1.0f);
}

__global__ void deg_finish_kernel(float* deg, int n) {
    int i = blockIdx.x * blockDim.x + threadIdx.x;
    if (i < n) deg[i] = rsqrtf(deg[i]);   // deg -> d^{-1/2}
}

__global__ void edge_coef_kernel(const long long* __restrict__ src,
                                 const long long* __restrict__ dst,
                                 const float* __restrict__ dinv,
                                 float* __restrict__ coef, int e) {
    int i = blockIdx.x * blockDim.x + threadIdx.x;
    if (i < e) coef[i] = dinv[(int)src[i]] * dinv[(int)dst[i]];
}

// ---------------------------------------------------------------------------
// FP32 WMMA GEMM: H[nrows x NO] = A[nrows x 64] * W[64 x NO],  NO compile-time.
// One wave per 16-row tile. A fragments (16 x v2f) cached in registers across
// all N tiles. Per N tile: load phase (16 B frags, immediate-offset loads off
// one base) then compute phase (16 back-to-back V_WMMA_F32_16X16X4_F32 with
// chained f32 accumulation).
// Layouts (ISA 7.12.2):
//   A 16x4: lane M = lane&15, K base = (lane>>4)*2, VGPR0/1 = K+0/K+1.
//   B 4x16: lane N = lane&15, K base = (lane>>4)*2 (transpose-symmetric).
//   C/D 16x16: VGPR r holds M = r + (lane>>4)*8, N = lane&15.
// ---------------------------------------------------------------------------
template <int NO>
__global__ void gemm_wmma_kernel(const float* __restrict__ A,
                                 const float* __restrict__ W,
                                 float* __restrict__ H,
                                 int ntiles) {
    int wid  = (blockIdx.x * blockDim.x + threadIdx.x) >> 5;
    if (wid >= ntiles) return;                 // wave-uniform: EXEC stays all-1s
    int lane = threadIdx.x & 31;
    int m    = lane & 15;
    int kh   = lane >> 4;                      // 0 or 1

    long long row0 = (long long)wid * 16;
    const float* arow = A + (row0 + m) * 64 + kh * 2;

    v2f af[16];
#pragma unroll
    for (int kk = 0; kk < 16; ++kk)
        af[kk] = *(const v2f*)(arow + kk * 4);

#pragma unroll
    for (int nt = 0; nt < NO / 16; ++nt) {
        // ---- load phase: 16 B fragments, constant offsets off one base ----
        const float* wcol = W + (kh * 2) * NO + nt * 16 + (lane & 15);
        v2f bf[16];
#pragma unroll
        for (int kk = 0; kk < 16; ++kk) {
            bf[kk].x = wcol[(kk * 4) * NO];
            bf[kk].y = wcol[(kk * 4 + 1) * NO];
        }
        // ---- compute phase: 16 chained WMMAs ----
        v8f acc = {};
#pragma unroll
        for (int kk = 0; kk < 16; ++kk)
            acc = __builtin_amdgcn_wmma_f32_16x16x4_f32(
                false, af[kk], false, bf[kk], (short)0, acc, false, false);

        float* hrow = H + (row0 + kh * 8) * NO + nt * 16 + (lane & 15);
#pragma unroll
        for (int r = 0; r < 8; ++r)
            hrow[r * NO] = acc[r];
    }
}

// ---------------------------------------------------------------------------
// agg[i,f] = h[i,f] * dinv[i]^2   (self-loop term; full overwrite = init)
// ---------------------------------------------------------------------------
__global__ void self_loop_kernel(const float* __restrict__ h,
                                 const float* __restrict__ dinv,
                                 float* __restrict__ agg,
                                 int n, int logF) {
    long long t = (long long)blockIdx.x * blockDim.x + threadIdx.x;
    if (t >= ((long long)n << logF)) return;
    int i = (int)(t >> logF);
    float d = dinv[i];
    agg[t] = h[t] * d * d;
}

// ---------------------------------------------------------------------------
// Edge scatter: one thread per (edge, feature); f32 atomics into L2-resident agg
// ---------------------------------------------------------------------------
__global__ void edge_scatter_kernel(const long long* __restrict__ src,
                                    const long long* __restrict__ dst,
                                    const float* __restrict__ coef,
                                    const float* __restrict__ h,
                                    float* __restrict__ agg,
                                    long long e, int logF) {
    long long t = (long long)blockIdx.x * blockDim.x + threadIdx.x;
    if (t >= (e << logF)) return;
    long long ei = t >> logF;
    int f = (int)(t & ((1 << logF) - 1));
    long long s = src[ei];
    long long d = dst[ei];
    float v = h[(s << logF) + f] * coef[ei];
    atomicAdd(&agg[(d << logF) + f], v);
}

// ---------------------------------------------------------------------------
// Fused bias + LayerNorm + ReLU, in place. One wave per node, 2 feats/lane.
// ---------------------------------------------------------------------------
__global__ void ln_relu_kernel(float* __restrict__ h,
                               const float* __restrict__ bias,
                               const float* __restrict__ g,
                               const float* __restrict__ b,
                               int n) {
    int node = (blockIdx.x * blockDim.x + threadIdx.x) >> 5;
    if (node >= n) return;
    int lane = threadIdx.x & 31;
    int f = lane * 2;
    float2 v = *(const float2*)&h[(long long)node * 64 + f];
    v.x += bias[f];
    v.y += bias[f + 1];
    float s  = v.x + v.y;
    float sq = v.x * v.x + v.y * v.y;
#pragma unroll
    for (int m = 16; m >= 1; m >>= 1) {
        s  += __shfl_xor(s,  m, 32);
        sq += __shfl_xor(sq, m, 32);
    }
    float mu  = s * (1.0f / 64.0f);
    float var = sq * (1.0f / 64.0f) - mu * mu;
    float rs  = rsqrtf(var + LN_EPS);
    float2 o;
    o.x = fmaxf(0.0f, (v.x - mu) * rs * g[f]     + b[f]);
    o.y = fmaxf(0.0f, (v.y - mu) * rs * g[f + 1] + b[f + 1]);
    *(float2*)&h[(long long)node * 64 + f] = o;
}

// ---------------------------------------------------------------------------
// Pooling
// ---------------------------------------------------------------------------
__global__ void zero_kernel(float* p, int n) {
    int i = blockIdx.x * blockDim.x + threadIdx.x;
    if (i < n) p[i] = 0.0f;
}

__global__ void count_kernel(const long long* __restrict__ batch, float* counts, int n) {
    int i = blockIdx.x * blockDim.x + threadIdx.x;
    if (i < n) atomicAdd(&counts[(int)batch[i]], 1.0f);
}

__global__ void pool_sum_kernel(const long long* __restrict__ batch,
                                const float* __restrict__ h,
                                float* __restrict__ sums, int n) {
    long long t = (long long)blockIdx.x * blockDim.x + threadIdx.x;
    if (t >= (long long)n * OUT_DIM) return;
    int i = (int)(t >> 7);              // OUT_DIM = 128
    int f = (int)(t & 127);
    atomicAdd(&sums[((int)batch[i]) * OUT_DIM + f], h[t]);
}

__global__ void pool_finalize_kernel(float* __restrict__ out,
                                     const float* __restrict__ counts,
                                     const float* __restrict__ b3) {
    int t = blockIdx.x * blockDim.x + threadIdx.x;
    if (t >= N_GRAPHS * OUT_DIM) return;
    int gidx = t >> 7;
    int f = t & 127;
    float c = fmaxf(counts[gidx], 1.0f);
    out[t] = out[t] / c + b3[f];        // mean(agg + b3) = mean(agg) + b3
}

// ---------------------------------------------------------------------------
// Launch
// ---------------------------------------------------------------------------
static inline int cdiv(long long a, long long b) { return (int)((a + b - 1) / b); }

extern "C" void kernel_launch(void* const* d_in, const int* in_sizes, int n_in,
                              void* d_out, int out_size, void* d_ws, size_t ws_size,
                              hipStream_t stream) {
    const float*     x    = (const float*)d_in[0];
    const long long* ei   = (const long long*)d_in[1];   // int64 [2, E]
    const long long* bat  = (const long long*)d_in[2];   // int64 [N]
    const float* W1 = (const float*)d_in[3];
    const float* b1 = (const float*)d_in[4];
    const float* g1 = (const float*)d_in[5];
    const float* be1= (const float*)d_in[6];
    const float* W2 = (const float*)d_in[7];
    const float* b2 = (const float*)d_in[8];
    const float* g2 = (const float*)d_in[9];
    const float* be2= (const float*)d_in[10];
    const float* W3 = (const float*)d_in[11];
    const float* b3 = (const float*)d_in[12];
    float* out = (float*)d_out;

    const long long N = N_NODES, E = N_EDGES;
    const long long* src = ei;
    const long long* dst = ei + E;

    // workspace layout (floats)
    float* w = (float*)d_ws;
    float* dinv   = w;                          // N
    float* coef   = w + 100096;                 // E
    float* bufA   = coef + N_EDGES;             // N*64
    float* bufB   = bufA + N_NODES * 64LL;      // N*64  (contiguous with bufA)
    float* bufC   = bufB + N_NODES * 64LL;      // N*128
    float* counts = bufC + N_NODES * 128LL;     // N_GRAPHS
    float* agg3   = bufA;                       // N*128 = bufA ∪ bufB

    const int B = 256;

    // degree -> d^{-1/2}, per-edge coefficients (shared across all 3 convs)
    init_deg_kernel<<<cdiv(N, B), B, 0, stream>>>(dinv, (int)N);
    deg_edges_kernel<<<cdiv(E, B), B, 0, stream>>>(dst, dinv, (int)E);
    deg_finish_kernel<<<cdiv(N, B), B, 0, stream>>>(dinv, (int)N);
    edge_coef_kernel<<<cdiv(E, B), B, 0, stream>>>(src, dst, dinv, coef, (int)E);

    const int ntiles = (int)(N / 16);           // 6250 (exact)
    const int gemmBlocks = cdiv(ntiles, 8);     // 8 waves/block

    // ---- Layer 1: h1 = x @ W1 -> bufA ; agg -> bufB ; LN+ReLU in place
    gemm_wmma_kernel<HID_DIM><<<gemmBlocks, B, 0, stream>>>(x, W1, bufA, ntiles);
    self_loop_kernel<<<cdiv(N * 64, B), B, 0, stream>>>(bufA, dinv, bufB, (int)N, 6);
    edge_scatter_kernel<<<cdiv(E * 64, B), B, 0, stream>>>(src, dst, coef, bufA, bufB, E, 6);
    ln_relu_kernel<<<cdiv(N * 32, B), B, 0, stream>>>(bufB, b1, g1, be1, (int)N);

    // ---- Layer 2: h2 = bufB @ W2 -> bufA ; agg -> bufB (gemm done, safe to reuse)
    gemm_wmma_kernel<HID_DIM><<<gemmBlocks, B, 0, stream>>>(bufB, W2, bufA, ntiles);
    self_loop_kernel<<<cdiv(N * 64, B), B, 0, stream>>>(bufA, dinv, bufB, (int)N, 6);
    edge_scatter_kernel<<<cdiv(E * 64, B), B, 0, stream>>>(src, dst, coef, bufA, bufB, E, 6);
    ln_relu_kernel<<<cdiv(N * 32, B), B, 0, stream>>>(bufB, b2, g2, be2, (int)N);

    // ---- Layer 3: h3 = bufB @ W3 -> bufC (128-wide) ; agg -> agg3 (bufA∪bufB)
    gemm_wmma_kernel<OUT_DIM><<<gemmBlocks, B, 0, stream>>>(bufB, W3, bufC, ntiles);
    self_loop_kernel<<<cdiv(N * 128, B), B, 0, stream>>>(bufC, dinv, agg3, (int)N, 7);
    edge_scatter_kernel<<<cdiv(E * 128, B), B, 0, stream>>>(src, dst, coef, bufC, agg3, E, 7);

    // ---- Global mean pool (b3 folded into finalize)
    zero_kernel<<<cdiv(N_GRAPHS * OUT_DIM, B), B, 0, stream>>>(out, N_GRAPHS * OUT_DIM);
    zero_kernel<<<1, 64, 0, stream>>>(counts, N_GRAPHS);
    count_kernel<<<cdiv(N, B), B, 0, stream>>>(bat, counts, (int)N);
    pool_sum_kernel<<<cdiv(N * 128, B), B, 0, stream>>>(bat, agg3, out, (int)N);
    pool_finalize_kernel<<<cdiv(N_GRAPHS * OUT_DIM, B), B, 0, stream>>>(out, counts, b3);
}